// EdgeSelector_87832081203320
// MI455X (gfx1250) — compile-verified
//
#include <hip/hip_runtime.h>
#include <math.h>

typedef __attribute__((ext_vector_type(2))) float v2f;
typedef __attribute__((ext_vector_type(4))) float v4f;
typedef __attribute__((ext_vector_type(8))) float v8f;

#define GAMMA   0.1f
#define EPS_    1e-8f
#define FEAT    256
#define BUDGET  64

// ---------------------------------------------------------------------------
// Phase 1: gather + cosine penalty + score. One wave = 16 candidates.
// K-relabeled V_WMMA_F32_16X16X4_F32: each lane feeds its own b128 quartet
// straight into A (no lane shuffles); B is built with the matching K labels
// from an LDS copy of the injected vector. Non-column-0 lanes read B from a
// zero-filled LDS region (no per-iteration masking VALU).
// Inner loop per 8 K: 1x global_load_b128 + 1x ds_load_b128 + 2x WMMA + 4 FMA.
// ---------------------------------------------------------------------------
__global__ __launch_bounds__(256) void edge_scores_wmma(
    const float* __restrict__ infl,      // [num_cand]
    const float* __restrict__ feats,     // [num_nodes, 256]
    const int*   __restrict__ cand_idx,  // [num_cand]
    const float* __restrict__ inj,       // [256]
    float* __restrict__ out_scores,      // d_out + BUDGET
    float* __restrict__ ws_scores,       // d_ws (mutable copy for top-k)
    int num_cand, int num_nodes)
{
    __shared__ __align__(16) float s_buf[2 * FEAT];  // [inj | zeros]
    __shared__ float lds_dot[128];                   // 8 waves * 16 candidates

    const int tid  = threadIdx.x;
    const int lane = tid & 31;
    const int wave = tid >> 5;
    const int m    = lane & 15;          // candidate slot (M row)
    const int half = lane >> 4;          // K-quartet owner within lane pair

    // stage injected vector + zero pad (16 B per thread, first 128 threads)
    if (tid < FEAT / 4) {
        *(v4f*)&s_buf[tid * 4] = *(const v4f*)&inj[tid * 4];
    } else if (tid < FEAT / 2) {
        v4f z = {0.0f, 0.0f, 0.0f, 0.0f};
        *(v4f*)&s_buf[FEAT + (tid - FEAT / 4) * 4] = z;
    }
    __syncthreads();

    // ||inj||^2 once: 8 floats/lane + 5-step butterfly
    float injn;
    {
        v4f j0 = *(const v4f*)&s_buf[lane * 8];
        v4f j1 = *(const v4f*)&s_buf[lane * 8 + 4];
        injn = j0.x * j0.x + j0.y * j0.y + j0.z * j0.z + j0.w * j0.w
             + j1.x * j1.x + j1.y * j1.y + j1.z * j1.z + j1.w * j1.w;
        #pragma unroll
        for (int off = 16; off >= 1; off >>= 1)
            injn += __shfl_xor(injn, off, 32);
    }

    const int cbase = blockIdx.x * 128 + wave * 16;
    const int c     = cbase + m;
    const int cc    = (c < num_cand) ? c : (num_cand - 1);   // clamp: keep EXEC full for WMMA
    int row = cand_idx[cc];
    row = (row < 0) ? 0 : ((row >= num_nodes) ? (num_nodes - 1) : row);

    // this lane's quartet base: actual K = kb + 4*half .. +3
    const float* rp = feats + (size_t)row * FEAT + 4 * half;
    // B source: injected chunk for column-0 lanes (m==0), zeros otherwise
    const float* bp = &s_buf[((m == 0) ? 0 : FEAT) + 4 * half];

    v8f   acc = {};
    float nrm = 0.0f;                    // partial ||cand||^2 (K = 4h..4h+3 mod 8)

    #pragma unroll 2
    for (int kb = 0; kb < FEAT; kb += 8) {
        v4f q = *(const v4f*)(rp + kb);  // feature quartet (global, b128)
        v4f i = *(const v4f*)(bp + kb);  // matching inj quartet / zeros (LDS, b128)

        nrm = fmaf(q.x, q.x, nrm);
        nrm = fmaf(q.y, q.y, nrm);
        nrm = fmaf(q.z, q.z, nrm);
        nrm = fmaf(q.w, q.w, nrm);

        // K-relabeled operands: A/B share the labeling, so the dot is exact.
        v2f a0 = {q.x, q.y}, b0 = {i.x, i.y};  // covers K = kb,kb+1,kb+4,kb+5
        v2f a1 = {q.z, q.w}, b1 = {i.z, i.w};  // covers K = kb+2,kb+3,kb+6,kb+7

        acc = __builtin_amdgcn_wmma_f32_16x16x4_f32(
                  false, a0, false, b0, (short)0, acc, false, false);
        acc = __builtin_amdgcn_wmma_f32_16x16x4_f32(
                  false, a1, false, b1, (short)0, acc, false, false);
    }

    // combine K-halves (lanes m <-> m+16)
    const float nrm2 = nrm + __shfl_xor(nrm, 16, 32);

    // D column 0: lane 0 holds M=0..7 in acc[0..7], lane 16 holds M=8..15
    if (m == 0) {
        const int mb = half * 8;
        #pragma unroll
        for (int j = 0; j < 8; ++j)
            lds_dot[wave * 16 + mb + j] = acc[j];
    }
    __syncthreads();
    const float dot = lds_dot[wave * 16 + m];

    if (half == 0 && c < num_cand) {
        const float cn = fmaxf(sqrtf(nrm2), EPS_);
        const float in = fmaxf(sqrtf(injn), EPS_);
        const float score = infl[c] - GAMMA * (dot / (cn * in));
        out_scores[c] = score;
        ws_scores[c]  = score;   // fully rewritten every call -> replay-safe
    }
}

// ---------------------------------------------------------------------------
// Phase 2: top-64 via 64 rounds of block-wide argmax over L2-resident scores.
// Smallest-index tie-break matches jax.lax.top_k stability.
// ---------------------------------------------------------------------------
__global__ __launch_bounds__(1024) void topk64_kernel(
    const int* __restrict__ cand_idx,
    float* __restrict__ ws_scores,       // mutated: winners set to -inf
    float* __restrict__ out_sel,         // d_out[0..63] (ids as float, exact < 2^24)
    int num_cand)
{
    __shared__ float sv[1024];
    __shared__ int   si[1024];
    const int tid = threadIdx.x;

    for (int t = 0; t < BUDGET; ++t) {
        float best = -INFINITY;
        int   bi   = -1;
        for (int i = tid; i < num_cand; i += 1024) {
            const float v = ws_scores[i];
            if (v > best) { best = v; bi = i; }   // strict > keeps smallest i
        }
        sv[tid] = best;
        si[tid] = bi;
        __syncthreads();

        for (int s = 512; s > 0; s >>= 1) {
            if (tid < s) {
                const float ov = sv[tid + s];
                const int   oi = si[tid + s];
                if (oi >= 0 &&
                    (ov > sv[tid] ||
                     (ov == sv[tid] && (si[tid] < 0 || oi < si[tid])))) {
                    sv[tid] = ov;
                    si[tid] = oi;
                }
            }
            __syncthreads();
        }

        if (tid == 0) {
            const int j = si[0];
            if (j >= 0) {
                out_sel[t]   = (float)cand_idx[j];
                ws_scores[j] = -INFINITY;
            } else {
                out_sel[t] = 0.0f;
            }
        }
        __syncthreads();
    }
}

// ---------------------------------------------------------------------------
extern "C" void kernel_launch(void* const* d_in, const int* in_sizes, int n_in,
                              void* d_out, int out_size, void* d_ws, size_t ws_size,
                              hipStream_t stream)
{
    const float* infl  = (const float*)d_in[0];   // influence_scores [num_cand]
    const float* feats = (const float*)d_in[1];   // node_features [num_nodes,256]
    const int*   cand  = (const int*)  d_in[2];   // candidate_indices [num_cand]
    const float* inj   = (const float*)d_in[3];   // injected_feat [256]

    const int num_cand  = in_sizes[0];
    const int num_nodes = in_sizes[1] / FEAT;

    float* out        = (float*)d_out;
    float* out_scores = out + BUDGET;             // scores after 64 selected ids
    float* ws_scores  = (float*)d_ws;

    const int blocks = (num_cand + 127) / 128;    // 128 candidates / 256-thread block
    edge_scores_wmma<<<blocks, 256, 0, stream>>>(
        infl, feats, cand, inj, out_scores, ws_scores, num_cand, num_nodes);

    topk64_kernel<<<1, 1024, 0, stream>>>(cand, ws_scores, out, num_cand);
}